// WaveInterferenceDecoder_31215822307556
// MI455X (gfx1250) — compile-verified
//
#include <hip/hip_runtime.h>
#include <cstdint>
#include <cstddef>

#define HEADS      8
#define HDIM       64      // H = HEADS * HPH
#define BLTOT      128     // B * L
#define VOCAB_N    32000
#define BLPB       32      // bl positions per block (grid.y = 4)
#define THREADS    256

// ---------------- CDNA5 async global->LDS staging (guarded) ----------------
#if defined(__gfx1250__) && __has_builtin(__builtin_amdgcn_global_load_async_to_lds_b32)
#define USE_ASYNC_LDS 1
#else
#define USE_ASYNC_LDS 0
#endif

#if USE_ASYNC_LDS
typedef __attribute__((address_space(1))) int GInt;   // global (device) int*
typedef __attribute__((address_space(3))) int LInt;   // LDS (shared) int*
#endif

__device__ __forceinline__ void stage_b32(const float* gsrc, float* ldst) {
#if USE_ASYNC_LDS
  __builtin_amdgcn_global_load_async_to_lds_b32(
      (GInt*)(uintptr_t)gsrc, (LInt*)(uintptr_t)ldst, 0, 0);
#else
  *ldst = *gsrc;
#endif
}

__device__ __forceinline__ void stage_wait() {
#if USE_ASYNC_LDS
#if __has_builtin(__builtin_amdgcn_s_wait_asynccnt)
  __builtin_amdgcn_s_wait_asynccnt(0);
#else
  asm volatile("s_wait_asynccnt 0" ::: "memory");
#endif
#endif
}

// ---------------- Kernel 1: per-position precompute ------------------------
// ws layout: [0,8192) = cw = W[h]*a*cos(p - ip)
//            [8192,16384) = exp(f)
//            [16384,24576) = exp(-f)
__global__ __launch_bounds__(THREADS)
void wave_prep_kernel(const float* __restrict__ freqs,
                      const float* __restrict__ amps,
                      const float* __restrict__ phases,
                      const float* __restrict__ iph,   // (HEADS*HPH) = 64
                      const float* __restrict__ W,     // (HEADS) = 8
                      float* __restrict__ ws) {
  const int t = blockIdx.x * THREADS + threadIdx.x;   // 0..8191
  if (t >= BLTOT * HDIM) return;
  const int j = t & (HDIM - 1);                        // h*8 + k
  const float f = freqs[t];
  const float c = amps[t] * __cosf(phases[t] - iph[j]);
  ws[t]                    = c * W[j >> 3];
  ws[BLTOT * HDIM + t]     = __expf(f);
  ws[2 * BLTOT * HDIM + t] = __expf(-f);
}

// ---------------- Kernel 2: main decode ------------------------------------
// exp(-|f - g|) == min( e^f * e^-g , e^-f * e^g )  -> no exp in the hot loop.
__global__ __launch_bounds__(THREADS)
void wave_decoder_main_kernel(const float* __restrict__ vp,   // (VOCAB, 64)
                              const float* __restrict__ ws,
                              const float* __restrict__ bptr,
                              float* __restrict__ out) {      // (128, VOCAB)
  __shared__ float cwS [BLPB * HDIM];
  __shared__ float efS [BLPB * HDIM];
  __shared__ float emfS[BLPB * HDIM];

  const int tid    = threadIdx.x;
  const int blBase = blockIdx.y * BLPB;
  const float* cwG  = ws + blBase * HDIM;
  const float* efG  = ws + BLTOT * HDIM + blBase * HDIM;
  const float* emfG = ws + 2 * BLTOT * HDIM + blBase * HDIM;

  // Stage this block's 24KB slice of uniforms into LDS (async on gfx1250).
  for (int i = tid; i < BLPB * HDIM; i += THREADS) {
    stage_b32(cwG + i,  cwS + i);
    stage_b32(efG + i,  efS + i);
    stage_b32(emfG + i, emfS + i);
  }
  stage_wait();
  __syncthreads();

  const int v = blockIdx.x * THREADS + tid;             // 125*256 == 32000
  const float* __restrict__ vrow = vp + (size_t)v * HDIM;
  __builtin_prefetch(vrow, 0, 0);
  __builtin_prefetch(vrow + 32, 0, 0);

  const float bias = bptr[0];
  float acc[BLPB];
#pragma unroll
  for (int i = 0; i < BLPB; ++i) acc[i] = bias;

#pragma unroll 1
  for (int jc = 0; jc < 4; ++jc) {                      // j in chunks of 16
    // vocab-side exps: computed exactly once per (thread, j)
    float ev[16], emv[16];
    const float4* v4 = (const float4*)(vrow + jc * 16);
#pragma unroll
    for (int q = 0; q < 4; ++q) {
      const float4 x = v4[q];
      ev[4 * q + 0] = __expf(x.x);  emv[4 * q + 0] = __expf(-x.x);
      ev[4 * q + 1] = __expf(x.y);  emv[4 * q + 1] = __expf(-x.y);
      ev[4 * q + 2] = __expf(x.z);  emv[4 * q + 2] = __expf(-x.z);
      ev[4 * q + 3] = __expf(x.w);  emv[4 * q + 3] = __expf(-x.w);
    }
#pragma unroll
    for (int bl = 0; bl < BLPB; ++bl) {
      const float4* cw4 = (const float4*)&cwS [bl * HDIM + jc * 16];
      const float4* ef4 = (const float4*)&efS [bl * HDIM + jc * 16];
      const float4* em4 = (const float4*)&emfS[bl * HDIM + jc * 16];
      float a = acc[bl];
#pragma unroll
      for (int q = 0; q < 4; ++q) {
        const float4 cw = cw4[q];                       // ds_load_b128 broadcasts
        const float4 ef = ef4[q];
        const float4 em = em4[q];
        a = fmaf(cw.x, fminf(ef.x * emv[4 * q + 0], em.x * ev[4 * q + 0]), a);
        a = fmaf(cw.y, fminf(ef.y * emv[4 * q + 1], em.y * ev[4 * q + 1]), a);
        a = fmaf(cw.z, fminf(ef.z * emv[4 * q + 2], em.z * ev[4 * q + 2]), a);
        a = fmaf(cw.w, fminf(ef.w * emv[4 * q + 3], em.w * ev[4 * q + 3]), a);
      }
      acc[bl] = a;
    }
  }

#pragma unroll
  for (int bl = 0; bl < BLPB; ++bl) {
    out[(size_t)(blBase + bl) * VOCAB_N + v] = acc[bl]; // coalesced along v
  }
}

// ---------------- launch ----------------------------------------------------
extern "C" void kernel_launch(void* const* d_in, const int* in_sizes, int n_in,
                              void* d_out, int out_size, void* d_ws, size_t ws_size,
                              hipStream_t stream) {
  const float* freqs  = (const float*)d_in[0];
  const float* amps   = (const float*)d_in[1];
  const float* phases = (const float*)d_in[2];
  const float* vp     = (const float*)d_in[3];
  const float* iph    = (const float*)d_in[4];
  const float* W      = (const float*)d_in[5];
  const float* b      = (const float*)d_in[6];
  float* out = (float*)d_out;
  float* ws  = (float*)d_ws;   // needs 3*128*64*4 = 96KB

  wave_prep_kernel<<<dim3((BLTOT * HDIM + THREADS - 1) / THREADS), THREADS, 0, stream>>>(
      freqs, amps, phases, iph, W, ws);

  dim3 grid(VOCAB_N / THREADS, BLTOT / BLPB);           // (125, 4)
  wave_decoder_main_kernel<<<grid, THREADS, 0, stream>>>(vp, ws, b, out);
}